// Group_38371237822530
// MI455X (gfx1250) — compile-verified
//
#include <hip/hip_runtime.h>

// Problem constants (from reference): x [B=8, N=16384, C=13], 512 groups of 32.
#define NB        8
#define N_PTS     16384
#define N_CH      13
#define N_GROUP   512
#define GS        32                       // neighbors per group (k)
#define FPS_THREADS 1024
#define PPT       (N_PTS / FPS_THREADS)    // 16 points per thread in FPS

#define TILE_C    16                       // centers per workgroup in KNN
#define K2_WAVES  16
#define K2_THREADS (K2_WAVES * 32)         // 512
#define CHUNK_COLS (K2_WAVES * 16)         // 256 points per chunk
#define N_CHUNKS  (N_PTS / CHUNK_COLS)     // 64

typedef __attribute__((ext_vector_type(2))) float v2f;
typedef __attribute__((ext_vector_type(8))) float v8f;
typedef unsigned long long u64;
typedef unsigned int       u32;

// Order-preserving float->u32 transform (works for negative scores too).
__device__ __forceinline__ u32 fkey(float f) {
    u32 u = __float_as_uint(f);
    u ^= (u32)(((int)u >> 31)) | 0x80000000u;   // neg: flip all; pos: flip sign
    return u;
}

__device__ __forceinline__ u64 shfl_xor_u64(u64 v, int m) {
    int lo = (int)(u32)v, hi = (int)(u32)(v >> 32);
    lo = __shfl_xor(lo, m, 32);
    hi = __shfl_xor(hi, m, 32);
    return ((u64)(u32)hi << 32) | (u32)lo;
}

// ---------------------------------------------------------------------------
// Kernel 1: farthest point sampling. One workgroup per batch. The whole xyz
// cloud (192 KB) lives in LDS (CDNA5 WGP has 320 KB) so the 511 sequential
// sweeps never touch memory. min_d kept in registers (16 per thread).
// ---------------------------------------------------------------------------
__global__ __launch_bounds__(FPS_THREADS)
void fps_kernel(const float* __restrict__ x,
                float* __restrict__ out_centers,   // d_out tail  [B,512,3]
                float* __restrict__ ws_centers)    // workspace   [B,512,3]
{
    extern __shared__ float dynsm[];
    float* sx = dynsm;
    float* sy = dynsm + N_PTS;
    float* sz = dynsm + 2 * N_PTS;
    __shared__ float red_v[32];
    __shared__ int   red_i[32];
    __shared__ int   s_bidx;

    const int b = blockIdx.x;
    const int t = threadIdx.x;
    const int lane = t & 31;
    const int wid  = t >> 5;
    const float* xb = x + (size_t)b * N_PTS * N_CH;

    // stage xyz (channels 4..6) into LDS
    for (int j = 0; j < PPT; ++j) {
        int p = t + j * FPS_THREADS;
        const float* src = xb + (size_t)p * N_CH + 4;
        sx[p] = src[0]; sy[p] = src[1]; sz[p] = src[2];
    }
    __syncthreads();

    float md[PPT];
#pragma unroll
    for (int j = 0; j < PPT; ++j) md[j] = 1.0e10f;   // matches reference init

    int last = 0;
    if (t == 0) {   // deterministic start at point 0
        float cx = sx[0], cy = sy[0], cz = sz[0];
        float* oc = out_centers + (size_t)b * N_GROUP * 3;
        float* wc = ws_centers  + (size_t)b * N_GROUP * 3;
        oc[0] = cx; oc[1] = cy; oc[2] = cz;
        wc[0] = cx; wc[1] = cy; wc[2] = cz;
    }

    for (int g = 1; g < N_GROUP; ++g) {
        const float lx = sx[last], ly = sy[last], lz = sz[last];
        float bv = -1.0f; int bi = 0;
#pragma unroll
        for (int j = 0; j < PPT; ++j) {
            int p = t + j * FPS_THREADS;
            float dx = sx[p] - lx, dy = sy[p] - ly, dz = sz[p] - lz;
            float d = dx * dx + dy * dy + dz * dz;
            float m = fminf(md[j], d);
            md[j] = m;
            if (m > bv) { bv = m; bi = p; }    // strict > keeps first (lowest p)
        }
        // wave32 argmax (ties -> lower index, matching jnp.argmax)
#pragma unroll
        for (int off = 16; off > 0; off >>= 1) {
            float ov = __shfl_down(bv, off, 32);
            int   oi = __shfl_down(bi, off, 32);
            if (ov > bv || (ov == bv && oi < bi)) { bv = ov; bi = oi; }
        }
        if (lane == 0) { red_v[wid] = bv; red_i[wid] = bi; }
        __syncthreads();
        if (wid == 0) {
            bv = red_v[lane]; bi = red_i[lane];
#pragma unroll
            for (int off = 16; off > 0; off >>= 1) {
                float ov = __shfl_down(bv, off, 32);
                int   oi = __shfl_down(bi, off, 32);
                if (ov > bv || (ov == bv && oi < bi)) { bv = ov; bi = oi; }
            }
            if (lane == 0) {
                s_bidx = bi;
                float cx = sx[bi], cy = sy[bi], cz = sz[bi];
                float* oc = out_centers + ((size_t)b * N_GROUP + g) * 3;
                float* wc = ws_centers  + ((size_t)b * N_GROUP + g) * 3;
                oc[0] = cx; oc[1] = cy; oc[2] = cz;
                wc[0] = cx; wc[1] = cy; wc[2] = cz;
            }
        }
        __syncthreads();
        last = s_bidx;
    }
}

// ---------------------------------------------------------------------------
// Kernel 2: KNN (top-32 smallest d2) + gather, distance matrix on WMMA.
// One workgroup per (batch, tile of 16 centers): grid = 8*32 = 256, 16 waves.
// Score s[m][n] = |p_n|^2 - 2 c_m . p_n  via V_WMMA_F32_16X16X4_F32:
//   A[16x4] = (-2cx, -2cy, -2cz, 0), B[4x16] = (px,py,pz,0)^T, C = |p|^2.
// Selection uses packed u64 keys (fkey(score)<<32 | idx): branchless 32-step
// compare-swap insertion per candidate, single-u64 wave merge at the end.
// ---------------------------------------------------------------------------
__global__ __launch_bounds__(K2_THREADS)
void knn_gather_kernel(const float* __restrict__ x,
                       const float* __restrict__ ws_centers,
                       float* __restrict__ out_neigh)
{
    __shared__ float s_scores[TILE_C][CHUNK_COLS];   // 16 KB
    __shared__ int   s_sel[TILE_C][GS];              // 2 KB

    const int b    = blockIdx.x >> 5;                // 32 center-tiles per batch
    const int g0   = (blockIdx.x & 31) * TILE_C;
    const int t    = threadIdx.x;
    const int lane = t & 31;
    const int w    = t >> 5;                         // wave id
    const int col  = lane & 15;
    const bool hi  = lane >= 16;

    const float* xb = x + (size_t)b * N_PTS * N_CH;
    const float* cb = ws_centers + ((size_t)b * N_GROUP + g0) * 3;

    // A matrix per-lane values (ISA 32-bit A 16x4 layout:
    // lanes 0-15 -> K=0,1 ; lanes 16-31 -> K=2,3, M = lane%16)
    const float cx = cb[col * 3 + 0];
    const float cy = cb[col * 3 + 1];
    const float cz = cb[col * 3 + 2];
    v2f a;
    a.x = hi ? (-2.0f * cz) : (-2.0f * cx);
    a.y = hi ? 0.0f         : (-2.0f * cy);

    // per-lane sorted (ascending) top-32 keys: (fkey(score)<<32)|idx
    u64 best[GS];
#pragma unroll
    for (int i = 0; i < GS; ++i) best[i] = ~0ull;

    for (int ch = 0; ch < N_CHUNKS; ++ch) {
        const int chunk_base = ch * CHUNK_COLS;

        // ---- 16x16 score tile via WMMA (uniform control flow, EXEC all 1s)
        const int p = chunk_base + w * 16 + col;
        const float* ps = xb + (size_t)p * N_CH + 4;
        const float px = ps[0], py = ps[1], pz = ps[2];
        const float pn = px * px + py * py + pz * pz;
        // B 4x16 layout: V0: lanes0-15 K=0(x), lanes16-31 K=1(y);
        //                V1: lanes0-15 K=2(z), lanes16-31 K=3(0)
        v2f bm;
        bm.x = hi ? py   : px;
        bm.y = hi ? 0.0f : pz;
        v8f c;
#pragma unroll
        for (int r = 0; r < 8; ++r) c[r] = pn;      // |p|^2 (column-constant)
        c = __builtin_amdgcn_wmma_f32_16x16x4_f32(
                /*neg_a=*/false, a, /*neg_b=*/false, bm,
                /*c_mod=*/(short)0, c, /*reuse_a=*/false, /*reuse_b=*/false);

        // transpose tile through LDS: D layout VGPR r, lane l ->
        // row = (l<16 ? r : r+8), col = l%16
#pragma unroll
        for (int r = 0; r < 8; ++r) {
            const int row = hi ? (r + 8) : r;
            s_scores[row][w * 16 + col] = c[r];
        }
        __syncthreads();

        // ---- selection: wave w owns center row w; 8 candidates per lane
        for (int i = 0; i < CHUNK_COLS / 32; ++i) {
            const int   cc  = lane + i * 32;
            const float v   = s_scores[w][cc];
            const u64   key = ((u64)fkey(v) << 32) | (u32)(chunk_base + cc);
            if (key < best[GS - 1]) {
                u64 cur = key;
#pragma unroll
                for (int q = 0; q < GS; ++q) {       // branchless cmp-swap chain
                    const u64 bq = best[q];
                    const bool lt = cur < bq;
                    best[q] = lt ? cur : bq;
                    cur     = lt ? bq  : cur;
                }
            }
        }
        __syncthreads();
    }

    // ---- merge 32 per-lane lists -> row top-32, emitted in ascending d2
    for (int tsel = 0; tsel < GS; ++tsel) {
        const u64 mykey = best[0];
        u64 m = mykey;
#pragma unroll
        for (int off = 16; off > 0; off >>= 1) {     // wave min over u64 keys
            const u64 o = shfl_xor_u64(m, off);
            m = (o < m) ? o : m;
        }
        if (lane == 0) s_sel[w][tsel] = (int)(u32)m;
        if (mykey == m) {                             // unique winner pops
#pragma unroll
            for (int q = 0; q < GS - 1; ++q) best[q] = best[q + 1];
            best[GS - 1] = ~0ull;
        }
    }
    __syncthreads();

    // ---- gather: 16 rows x 32 neighbors, 13 channels, center-relative xyz
    {
        const int row = w;
        const int k   = lane;
        const int idx = s_sel[row][k];
        const float* src = xb + (size_t)idx * N_CH;
        const float ccx = cb[row * 3 + 0];
        const float ccy = cb[row * 3 + 1];
        const float ccz = cb[row * 3 + 2];
        float* dst = out_neigh +
            (((size_t)(b * N_GROUP + g0 + row)) * GS + k) * N_CH;
#pragma unroll
        for (int cc = 0; cc < N_CH; ++cc) {
            float v = src[cc];
            if (cc == 4) v -= ccx;
            if (cc == 5) v -= ccy;
            if (cc == 6) v -= ccz;
            dst[cc] = v;
        }
    }
}

// ---------------------------------------------------------------------------
extern "C" void kernel_launch(void* const* d_in, const int* in_sizes, int n_in,
                              void* d_out, int out_size, void* d_ws, size_t ws_size,
                              hipStream_t stream) {
    (void)in_sizes; (void)n_in; (void)out_size; (void)ws_size;
    const float* x = (const float*)d_in[0];
    float* out        = (float*)d_out;
    float* out_neigh  = out;                                        // [8,512,32,13]
    float* out_center = out + (size_t)NB * N_GROUP * GS * N_CH;     // [8,512,3]
    float* ws_centers = (float*)d_ws;                               // [8,512,3]

    // FPS: one WG per batch, 192 KB dynamic LDS for the xyz cloud.
    fps_kernel<<<dim3(NB), dim3(FPS_THREADS),
                 3 * N_PTS * sizeof(float), stream>>>(x, out_center, ws_centers);

    // KNN + gather: 256 WGs x 512 threads, WMMA distance matrix.
    knn_gather_kernel<<<dim3(NB * (N_GROUP / TILE_C)), dim3(K2_THREADS),
                        0, stream>>>(x, ws_centers, out_neigh);
}